// Criterion_69776038691135
// MI455X (gfx1250) — compile-verified
//
#include <hip/hip_runtime.h>
#include <hip/hip_bf16.h>
#include <math.h>

typedef __attribute__((ext_vector_type(2))) float v2f;
typedef __attribute__((ext_vector_type(8))) float v8f;

#define Q 256
#define NROW 4096      // 64*64
#define PER_T 16       // NROW / 256
#define NEG_FILL -1000000.0f
#define EOS_COEF 0.25f

// Deterministic 256-value block sum using V_WMMA_F32_16X16X4_F32.
// All 256 threads call this (EXEC all ones per wave, as WMMA requires).
// Layout per ISA 7.12.2:
//   A (16x4 f32): lanes 0-15 hold M=lane with VGPR0=K0, VGPR1=K1;
//                 lanes 16-31 hold M=lane-16 with VGPR0=K2, VGPR1=K3.
//   D (16x16 f32): VGPR r, lanes 0-15 -> D[M=r][N=lane]; lanes 16-31 -> D[M=r+8].
// With B = ones, D[m][*] = rowsum_m. Per-lane sum of 8 VGPRs gives
// sum(rows 0-7) in lanes 0-15 and sum(rows 8-15) in lanes 16-31; one
// shfl_xor(16) completes the 64-element chunk sum; 4 chained WMMAs cover 256.
__device__ inline float block_sum_wmma(float v, float* s_red, int tid) {
  s_red[tid] = v;
  __syncthreads();
  const int lane = tid & 31;
  v8f acc = {0.f, 0.f, 0.f, 0.f, 0.f, 0.f, 0.f, 0.f};
  v2f ones = {1.0f, 1.0f};
#pragma unroll
  for (int chunk = 0; chunk < 4; ++chunk) {
    const int base = chunk * 64;
    const int idx = (lane < 16) ? (base + lane * 4) : (base + (lane - 16) * 4 + 2);
    v2f a;
    a[0] = s_red[idx];
    a[1] = s_red[idx + 1];
    acc = __builtin_amdgcn_wmma_f32_16x16x4_f32(
        /*neg_a=*/false, a, /*neg_b=*/false, ones,
        /*c_mod=*/(short)0, acc, /*reuse_a=*/false, /*reuse_b=*/false);
  }
  float s = acc[0] + acc[1] + acc[2] + acc[3] + acc[4] + acc[5] + acc[6] + acc[7];
  s += __shfl_xor(s, 16, 32);   // combine the two 8-row halves -> total in all lanes
  __syncthreads();              // s_red safe to reuse after this
  return s;
}

__global__ __launch_bounds__(256) void criterion_row_kernel(
    const float* __restrict__ predL, const float* __restrict__ predQv,
    const float* __restrict__ tgtL,  const float* __restrict__ tgtQv,
    float* __restrict__ ws_bce, float* __restrict__ ws_ce, float* __restrict__ ws_mse)
{
  __shared__ int   s_scan[256];
  __shared__ float s_lined[256];
  __shared__ float s_match[256];
  __shared__ float s_red[256];

  const int b = blockIdx.x;
  const int tid = threadIdx.x;

  s_lined[tid] = NEG_FILL;
  s_match[tid] = 0.0f;

  // ---- stream 16 contiguous cells per thread (b128 loads) ----
  const float4* p4 = (const float4*)(predL + (size_t)b * NROW + tid * PER_T);
  const float4* t4 = (const float4*)(tgtL  + (size_t)b * NROW + tid * PER_T);
  float pv[PER_T], tv[PER_T];
#pragma unroll
  for (int i = 0; i < 4; ++i) {
    float4 p = p4[i]; float4 t = t4[i];
    pv[4*i+0] = p.x; pv[4*i+1] = p.y; pv[4*i+2] = p.z; pv[4*i+3] = p.w;
    tv[4*i+0] = t.x; tv[4*i+1] = t.y; tv[4*i+2] = t.z; tv[4*i+3] = t.w;
  }

  // ---- masks + weighted BCE-with-logits ----
  int tmask = 0, pmask = 0;
  float bce = 0.0f;
#pragma unroll
  for (int j = 0; j < PER_T; ++j) {
    const float x = pv[j], t = tv[j];
    tmask |= (t == 1.0f) ? (1 << j) : 0;
    pmask |= (x > 0.0f)  ? (1 << j) : 0;
    const float w = (t == 0.0f) ? EOS_COEF : 1.0f;
    bce += w * (fmaxf(x, 0.0f) - x * t + log1pf(expf(-fabsf(x))));
  }

  // ---- block-wide exclusive scan of (t_count, p_count), packed 16/16 ----
  s_scan[tid] = (__popc(tmask) << 16) | __popc(pmask);
  __syncthreads();
  for (int off = 1; off < 256; off <<= 1) {
    const int v = s_scan[tid];
    const int u = (tid >= off) ? s_scan[tid - off] : 0;
    __syncthreads();
    s_scan[tid] = v + u;
    __syncthreads();
  }
  const int excl = (tid > 0) ? s_scan[tid - 1] : 0;

  // ---- rank walk + conflict-free scatter (t_ranks unique per row) ----
  int trk = excl >> 16;
  int prk = excl & 0xFFFF;
  const float* pq = predQv + (size_t)b * Q;
#pragma unroll
  for (int j = 0; j < PER_T; ++j) {
    const int tb = (tmask >> j) & 1;
    const int pb = (pmask >> j) & 1;
    if (tb && pb && prk < (Q - 1) && trk < Q) {
      s_lined[trk] = pq[prk];
      s_match[trk] = 1.0f;
    }
    trk += tb; prk += pb;
  }
  __syncthreads();

  // ---- per-class math: thread tid owns class c = tid ----
  const bool last = (tid == Q - 1);
  const float lined_c = s_lined[tid];                                 // classes < 255
  const float tq_c = last ? 0.0f : tgtQv[(size_t)b * Q + tid] * s_match[tid];

  // max over classes 0..254 (LDS tree, deterministic)
  s_red[tid] = last ? -3.0e38f : lined_c;
  __syncthreads();
  for (int off = 128; off > 0; off >>= 1) {
    if (tid < off) s_red[tid] = fmaxf(s_red[tid], s_red[tid + off]);
    __syncthreads();
  }
  const float m = s_red[0];
  __syncthreads();

  const float sumexp = block_sum_wmma(last ? 0.0f : expf(lined_c - m), s_red, tid);
  const float lse = m + logf(sumexp);
  const float tsum = block_sum_wmma(tq_c, s_red, tid);
  // unmatched classes: tq_c == 0 -> 0 * (NEG_FILL - lse) == 0 (finite), matches JAX
  const float num = block_sum_wmma(last ? 0.0f : tq_c * (lined_c - lse), s_red, tid);
  const float bce_row = block_sum_wmma(bce, s_red, tid);

  if (tid == 0) {
    ws_bce[b] = bce_row;
    ws_ce[b]  = -num / (tsum + 1e-10f);
    const float d = pq[Q - 1] - tgtQv[(size_t)b * Q + (Q - 1)];
    ws_mse[b] = d * d;   // lined[:,-1]=predQ[:,-1], match[:,-1]=1
  }
}

__global__ __launch_bounds__(256) void criterion_final_kernel(
    const float* __restrict__ ws_bce, const float* __restrict__ ws_ce,
    const float* __restrict__ ws_mse, float* __restrict__ out, int nB)
{
  __shared__ double s_red[256];
  const int tid = threadIdx.x;
  double bs = 0.0, cs = 0.0, ms = 0.0;
  for (int i = tid; i < nB; i += 256) {   // fixed strided order: deterministic
    bs += (double)ws_bce[i];
    cs += (double)ws_ce[i];
    ms += (double)ws_mse[i];
  }
  double vals[3] = {bs, cs, ms};
  double tot[3];
  for (int r = 0; r < 3; ++r) {
    s_red[tid] = vals[r];
    __syncthreads();
    for (int off = 128; off > 0; off >>= 1) {
      if (tid < off) s_red[tid] += s_red[tid + off];
      __syncthreads();
    }
    tot[r] = s_red[0];
    __syncthreads();
  }
  if (tid == 0) {
    const double nElem = (double)nB * (double)NROW;
    out[0] = (float)(tot[0] / nElem);                                   // loss_labels
    out[1] = (float)(200.0 * (tot[1] / 255.0) + tot[2] / (double)nB);   // loss_score
  }
}

extern "C" void kernel_launch(void* const* d_in, const int* in_sizes, int n_in,
                              void* d_out, int out_size, void* d_ws, size_t ws_size,
                              hipStream_t stream) {
  const float* predL = (const float*)d_in[0];  // [B,64,64]
  const float* predQ = (const float*)d_in[1];  // [B,256]
  const float* tgtL  = (const float*)d_in[2];  // [B,64,64]
  const float* tgtQ  = (const float*)d_in[3];  // [B,256]
  float* out = (float*)d_out;                  // 2 floats

  const int nB = in_sizes[1] / Q;              // 4096

  float* ws     = (float*)d_ws;
  float* ws_bce = ws;
  float* ws_ce  = ws + nB;
  float* ws_mse = ws + 2 * nB;

  criterion_row_kernel<<<nB, 256, 0, stream>>>(predL, predQ, tgtL, tgtQ,
                                               ws_bce, ws_ce, ws_mse);
  criterion_final_kernel<<<1, 256, 0, stream>>>(ws_bce, ws_ce, ws_mse, out, nB);
}